// OnlineEmbedding_64484638982170
// MI455X (gfx1250) — compile-verified
//
#include <hip/hip_runtime.h>
#include <stdint.h>

// Embedding gather: out[r,:] = table[ids[r],:], rows=819200, 64 f32 (256B) per row.
// Pure data movement -> gfx1250 async global<->LDS mover (ASYNCcnt path).
// Each wave32 moves 4 rows (2 x b128 async ops = 1KB) per direction per iteration,
// double-buffered in LDS. Output stores are non-temporal so the ~143MB unique
// working set of the table stays resident in the 192MB L2.

#define THREADS 256
#define WPB (THREADS / 32)   // waves per block

__global__ __launch_bounds__(THREADS) void embed_gather_async(
    const int* __restrict__ ids, const float* __restrict__ table,
    float* __restrict__ out, int rows)
{
    // Per wave: 2 sets x 1KB (each set = 2 halves x 512B = 32 lanes x 16B x 2)
    __shared__ char smem[WPB * 2048];

    const uint32_t tid   = threadIdx.x;
    const uint32_t lane  = tid & 31u;
    const uint32_t wib   = tid >> 5;     // wave index within block
    const uint32_t col16 = lane & 15u;   // 16B chunk within a 256B row
    const uint32_t sub   = lane >> 4;    // which row of a half (0/1)

    const uint64_t tab_base = (uint64_t)(uintptr_t)table;
    const uint64_t out_base = (uint64_t)(uintptr_t)out;

    // Generic->LDS: low 32 bits of the generic address are the LDS byte offset.
    const uint32_t lds_l = (uint32_t)(uintptr_t)smem + wib * 2048u + lane * 16u;

    const uint32_t quads  = (uint32_t)rows >> 2;                  // 4 rows per quad
    uint32_t       q      = blockIdx.x * WPB + wib;
    const uint32_t stride = gridDim.x * WPB;

    // Prologue: load quad q into set 0 (rows 4q+sub and 4q+2+sub).
    if (q < quads) {
        uint32_t r0 = q * 4u + sub;
        uint32_t g0 = (uint32_t)ids[r0] * 256u + col16 * 16u;
        uint32_t g1 = (uint32_t)ids[r0 + 2u] * 256u + col16 * 16u;
        asm volatile("global_load_async_to_lds_b128 %0, %1, %2"
                     :: "v"(lds_l), "v"(g0), "s"(tab_base) : "memory");
        asm volatile("global_load_async_to_lds_b128 %0, %1, %2"
                     :: "v"(lds_l + 512u), "v"(g1), "s"(tab_base) : "memory");
    }

    uint32_t sel = 0;
    while (q < quads) {
        uint32_t qn = q + stride;

        // Outstanding here: 2 loads (current set) + 2 stores (previous set).
        // Loads vs stores complete OOO, so only <=0 is a safe threshold.
        asm volatile("s_wait_asynccnt 0x0" ::: "memory");

        {   // Store current set. Row r = 4q + 2j + sub, so the global offset of
            // half j=1 is half j=0 plus 512B, and its LDS offset is also +512B:
            // IOFFSET is added to BOTH sides, so 'offset:512' reuses the operands.
            uint32_t ooff = q * 1024u + sub * 256u + col16 * 16u;
            uint32_t lcur = lds_l + sel * 1024u;
            asm volatile("global_store_async_from_lds_b128 %0, %1, %2 th:TH_STORE_NT"
                         :: "v"(ooff), "v"(lcur), "s"(out_base) : "memory");
            asm volatile("global_store_async_from_lds_b128 %0, %1, %2 offset:512 th:TH_STORE_NT"
                         :: "v"(ooff), "v"(lcur), "s"(out_base) : "memory");
        }

        if (qn < quads) {  // prefetch next quad into the other set
            uint32_t r0 = qn * 4u + sub;
            uint32_t g0 = (uint32_t)ids[r0] * 256u + col16 * 16u;
            uint32_t g1 = (uint32_t)ids[r0 + 2u] * 256u + col16 * 16u;
            uint32_t lnext = lds_l + (sel ^ 1u) * 1024u;
            asm volatile("global_load_async_to_lds_b128 %0, %1, %2"
                         :: "v"(lnext), "v"(g0), "s"(tab_base) : "memory");
            asm volatile("global_load_async_to_lds_b128 %0, %1, %2"
                         :: "v"(lnext + 512u), "v"(g1), "s"(tab_base) : "memory");
        }

        q = qn;
        sel ^= 1u;
    }

    // Drain outstanding stores (S_ENDPGM also implies wait-idle; be explicit).
    asm volatile("s_wait_asynccnt 0x0" ::: "memory");

    // Generality guard: rows % 4 trailing rows via plain VMEM (rows%4==0 here).
    const int tail0 = (int)(quads << 2);
    const int ntail = rows - tail0;
    if (ntail > 0 && blockIdx.x == 0) {
        for (int c = (int)tid; c < ntail * 16; c += THREADS) {
            int r = tail0 + (c >> 4);
            int k = c & 15;
            ((float4*)(out + (size_t)r * 64))[k] =
                ((const float4*)(table + (size_t)(uint32_t)ids[r] * 64))[k];
        }
    }
}

extern "C" void kernel_launch(void* const* d_in, const int* in_sizes, int n_in,
                              void* d_out, int out_size, void* d_ws, size_t ws_size,
                              hipStream_t stream) {
    const int*   ids   = (const int*)d_in[0];     // [BATCH*HIST] indices
    const float* table = (const float*)d_in[1];   // [VOCAB, 64] fp32
    float*       out   = (float*)d_out;           // [BATCH*HIST, 64] fp32

    const int rows = in_sizes[0];                 // 819200
    long long quads = (long long)(rows >> 2);     // 204800

    // 2048 blocks x 8 waves = 16K waves; ~12 grid-stride iterations per wave
    // (warms the double-buffer pipeline; 16K waves x 2KB in flight saturates HBM).
    int blocks = 2048;
    long long min_blocks = (quads + WPB - 1) / WPB;
    if ((long long)blocks > min_blocks) blocks = (int)(min_blocks > 0 ? min_blocks : 1);

    embed_gather_async<<<blocks, THREADS, 0, stream>>>(ids, table, out, rows);
}